// MMDropoutDiagonal_5970004542089
// MI455X (gfx1250) — compile-verified
//
#include <hip/hip_runtime.h>
#include <hip/hip_bf16.h>

// out[r, c] = x[r, c] * scale[c],  scale[c] = (u[c] >= P) ? vec[c] / (1-P) : 0
// Memory-bound streaming scale (1.07 GB traffic, ~46 us floor @ 23.3 TB/s).
// Non-temporal hints on the streamed loads/stores (CDNA5 TH field) since the
// dataset exceeds the 192 MB L2 and every element is touched exactly once.
//
// Grid-stride in float4-chunk units with stride = gridDim.x * 256: since
// blockDim.x == 256 == chunks-per-row, (i % 256) == threadIdx.x is invariant,
// so each thread's 4-wide dropout scale is computed once and reused forever.

typedef __attribute__((ext_vector_type(4))) float v4f;

#define DEPTH  1024
#define TPB    256            // == DEPTH/4 chunks per row; 8 wave32 waves
#define GRID   2048           // 64 rows per block on the 131072-row reference

__global__ __launch_bounds__(TPB, 1)
void MMDropoutDiagonal_5970004542089_kernel(const float* __restrict__ x,
                                            const float* __restrict__ vec,
                                            const float* __restrict__ u,
                                            float* __restrict__ out,
                                            unsigned n4)   // total float4 chunks
{
    const unsigned tid = threadIdx.x;

    // Per-thread dropout-scaled diagonal (loop-invariant; vec/u are 4 KB,
    // L2-resident across all blocks).
    const v4f v  = ((const v4f*)vec)[tid];
    const v4f uu = ((const v4f*)u)[tid];
    const float inv_keep = 1.0f / (1.0f - 0.1f);
    v4f s;
    s.x = (uu.x >= 0.1f) ? v.x * inv_keep : 0.0f;
    s.y = (uu.y >= 0.1f) ? v.y * inv_keep : 0.0f;
    s.z = (uu.z >= 0.1f) ? v.z * inv_keep : 0.0f;
    s.w = (uu.w >= 0.1f) ? v.w * inv_keep : 0.0f;

    const v4f* __restrict__ px = (const v4f*)x;
    v4f*       __restrict__ po = (v4f*)out;

    unsigned i            = blockIdx.x * (unsigned)TPB + tid;  // 32-bit index
    const unsigned stride = gridDim.x * (unsigned)TPB;         // ≡ 0 mod 256

    // 4-deep unroll: 4 independent 128-bit NT loads in flight, then 4 NT
    // stores. 32-bit offsets -> SADDR-base + scaled VGPR-offset addressing.
    for (; i + 3u * stride < n4; i += 4u * stride) {
        v4f a0 = __builtin_nontemporal_load(px + i);
        v4f a1 = __builtin_nontemporal_load(px + i + stride);
        v4f a2 = __builtin_nontemporal_load(px + i + 2u * stride);
        v4f a3 = __builtin_nontemporal_load(px + i + 3u * stride);
        a0 *= s;
        a1 *= s;
        a2 *= s;
        a3 *= s;
        __builtin_nontemporal_store(a0, po + i);
        __builtin_nontemporal_store(a1, po + i + stride);
        __builtin_nontemporal_store(a2, po + i + 2u * stride);
        __builtin_nontemporal_store(a3, po + i + 3u * stride);
    }
    for (; i < n4; i += stride) {
        v4f a = __builtin_nontemporal_load(px + i);
        a *= s;
        __builtin_nontemporal_store(a, po + i);
    }
}

extern "C" void kernel_launch(void* const* d_in, const int* in_sizes, int n_in,
                              void* d_out, int out_size, void* d_ws, size_t ws_size,
                              hipStream_t stream) {
    const float* x   = (const float*)d_in[0];   // [rows, 1024] f32
    const float* vec = (const float*)d_in[1];   // [1024] f32
    const float* u   = (const float*)d_in[2];   // [1024] f32
    float* out = (float*)d_out;                 // [rows, 1024] f32

    const unsigned n4 = (unsigned)(in_sizes[0] / 4);  // total float4 chunks

    hipLaunchKernelGGL(MMDropoutDiagonal_5970004542089_kernel,
                       dim3(GRID), dim3(TPB), 0, stream,
                       x, vec, u, out, n4);
}